// GraphLevelGNN_49795850830443
// MI455X (gfx1250) — compile-verified
//
#include <hip/hip_runtime.h>
#include <hip/hip_bf16.h>

// ---------------------------------------------------------------------------
// GCN 2-layer forward on gfx1250 (MI455X).
//   h1   = x @ W1                         (fp32 WMMA 16x16x4)
//   agg1 = D^-1/2 A D^-1/2 h1             (edge gather/scatter, f32 atomics)
//   z    = relu(agg1 + b1)
//   h2   = z @ W2                         (fp32 WMMA 16x16x4)
//   agg2 = D^-1/2 A D^-1/2 h2
//   out  = mean(agg2, axis=0) + b2        -> [1, 32]
// ---------------------------------------------------------------------------

typedef __attribute__((ext_vector_type(2))) float v2f;
typedef __attribute__((ext_vector_type(8))) float v8f;

// ---------------------------- elementwise helpers ---------------------------

__global__ void set_f32(float* __restrict__ p, float v, long long n) {
    long long i = (long long)blockIdx.x * blockDim.x + threadIdx.x;
    if (i < n) p[i] = v;
}

__global__ void zero_f32(float* __restrict__ p, long long n) {
    long long i = (long long)blockIdx.x * blockDim.x + threadIdx.x;
    if (i < n) p[i] = 0.0f;
}

// deg[dst[e]] += 1 for every real edge (self-loops pre-seeded as deg=1).
__global__ void count_deg(const long long* __restrict__ dstIdx,
                          float* __restrict__ deg, int nE) {
    int e = blockIdx.x * blockDim.x + threadIdx.x;
    if (e < nE) atomicAdd(&deg[(int)dstIdx[e]], 1.0f);
}

// In-place deg -> rsqrt(deg) (deg >= 1 always, guard anyway to match ref).
__global__ void deg_to_dinv(float* __restrict__ deg, int n) {
    int i = blockIdx.x * blockDim.x + threadIdx.x;
    if (i < n) {
        float d = deg[i];
        deg[i] = (d > 0.0f) ? rsqrtf(d) : 0.0f;
    }
}

__global__ void bias_relu64(const float* __restrict__ agg,
                            const float* __restrict__ b,
                            float* __restrict__ z, long long total) {
    long long i = (long long)blockIdx.x * blockDim.x + threadIdx.x;
    if (i < total) {
        float v = agg[i] + b[(int)(i & 63)];
        z[i] = v > 0.0f ? v : 0.0f;
    }
}

// ------------------------------ WMMA GEMM ----------------------------------
// H[nRows x FOUT] = A[nRows x FIN] * W[FIN x FOUT], fp32 accumulation.
// 256 threads = 8 waves; each wave computes 16 rows x FOUT cols via
// V_WMMA_F32_16X16X4_F32, K stepped by 4.
//
// W is staged in LDS pre-packed into the exact B-fragment layout
// [kblk][ntile][lane][2] so each fragment is one aligned ds_load_b64.
// A rows are clamped (not predicated) so fragment loads are branchless
// global_load_b64; out-of-range rows compute duplicates that the guarded
// store discards. WMMA sits in uniform control flow (EXEC all-1s).
template <int FIN, int FOUT>
__global__ __launch_bounds__(256) void gcn_gemm_wmma(
    const float* __restrict__ A, const float* __restrict__ W,
    float* __restrict__ H, int nRows) {
    constexpr int NT = FOUT / 16;  // output tiles along N
    constexpr int KB = FIN / 4;    // K blocks
    __shared__ float sW[KB * NT * 64];  // FIN*FOUT floats, fragment-packed

    const int tid  = threadIdx.x;
    const int wave = tid >> 5;
    const int lane = tid & 31;
    const int rowBase = blockIdx.x * 128 + wave * 16;

    // Cooperative pack of W into WMMA B-fragment layout:
    // fragment f = (kblk*NT + t); lane l holds W[kk][col], W[kk+1][col]
    // with kk = kblk*4 + (l>>4)*2, col = t*16 + (l&15).
    for (int i = tid; i < KB * NT * 32; i += 256) {
        int l    = i & 31;
        int f    = i >> 5;
        int t    = f % NT;
        int kblk = f / NT;
        int kk   = kblk * 4 + ((l >> 4) << 1);
        int col  = t * 16 + (l & 15);
        float2 wv;
        wv.x = W[(kk + 0) * FOUT + col];
        wv.y = W[(kk + 1) * FOUT + col];
        *(float2*)(&sW[(size_t)i * 2]) = wv;
    }
    __syncthreads();

    // Branchless A fragment source: clamp row, discard at store.
    const int mr = min(rowBase + (lane & 15), nRows - 1);
    const float* __restrict__ aRow =
        A + (size_t)mr * FIN + ((lane >> 4) << 1);  // +0 or +2 within K block

    v8f acc[NT];
#pragma unroll
    for (int t = 0; t < NT; ++t)
#pragma unroll
        for (int v = 0; v < 8; ++v) acc[t][v] = 0.0f;

    for (int kblk = 0; kblk < KB; ++kblk) {
        const float2 av = *(const float2*)(aRow + kblk * 4);
        v2f a;
        a.x = av.x;
        a.y = av.y;
#pragma unroll
        for (int t = 0; t < NT; ++t) {
            const float2 bv =
                *(const float2*)(&sW[((kblk * NT + t) << 6) + (lane << 1)]);
            v2f b;
            b.x = bv.x;
            b.y = bv.y;
            acc[t] = __builtin_amdgcn_wmma_f32_16x16x4_f32(
                false, a, false, b, (short)0, acc[t], false, false);
        }
    }

    // C/D layout: lane<16 -> col=lane, rows M=v ; lane>=16 -> col=lane-16, M=v+8
    const int rbase = rowBase + (lane >> 4) * 8;
    const int cbase = lane & 15;
#pragma unroll
    for (int t = 0; t < NT; ++t) {
#pragma unroll
        for (int v = 0; v < 8; ++v) {
            int row = rbase + v;
            if (row < nRows) H[(size_t)row * FOUT + t * 16 + cbase] = acc[t][v];
        }
    }
}

// --------------------------- edge scatter-add ------------------------------
// One thread per (edge, 4-feature chunk). agg[dst] += h[src] * dinv[s]*dinv[d]
template <int F>
__global__ void gcn_scatter_edges(const long long* __restrict__ srcIdx,
                                  const long long* __restrict__ dstIdx,
                                  const float* __restrict__ dinv,
                                  const float* __restrict__ h,
                                  float* __restrict__ agg, int nEdges) {
    constexpr int C  = F / 4;
    constexpr int SH = (F == 64) ? 4 : 3;
    unsigned long long idx =
        (unsigned long long)blockIdx.x * blockDim.x + threadIdx.x;
    int e = (int)(idx >> SH);
    int c = (int)(idx & (unsigned)(C - 1));
    if (e >= nEdges) return;
    int s = (int)srcIdx[e];
    int d = (int)dstIdx[e];
    float norm = dinv[s] * dinv[d];
    const float4 hv = *(const float4*)(h + (size_t)s * F + c * 4);
    float* o = agg + (size_t)d * F + c * 4;
    atomicAdd(o + 0, hv.x * norm);
    atomicAdd(o + 1, hv.y * norm);
    atomicAdd(o + 2, hv.z * norm);
    atomicAdd(o + 3, hv.w * norm);
}

// Self-loop contribution: agg[i] += h[i] * dinv[i]^2
template <int F>
__global__ void gcn_scatter_self(const float* __restrict__ dinv,
                                 const float* __restrict__ h,
                                 float* __restrict__ agg, int n) {
    constexpr int C  = F / 4;
    constexpr int SH = (F == 64) ? 4 : 3;
    unsigned long long idx =
        (unsigned long long)blockIdx.x * blockDim.x + threadIdx.x;
    int i = (int)(idx >> SH);
    int c = (int)(idx & (unsigned)(C - 1));
    if (i >= n) return;
    float dv = dinv[i];
    float norm = dv * dv;
    const float4 hv = *(const float4*)(h + (size_t)i * F + c * 4);
    float* o = agg + (size_t)i * F + c * 4;
    atomicAdd(o + 0, hv.x * norm);
    atomicAdd(o + 1, hv.y * norm);
    atomicAdd(o + 2, hv.z * norm);
    atomicAdd(o + 3, hv.w * norm);
}

// -------------------------- mean-pool reduction ----------------------------
// Block = 256 threads (8 node-rows x 32 feature-cols). Partial column sums
// accumulated in registers, reduced via LDS, atomically folded into csum[32].
__global__ __launch_bounds__(256) void colsum32(const float* __restrict__ agg2,
                                                float* __restrict__ csum,
                                                int n) {
    int c  = threadIdx.x & 31;
    int rl = threadIdx.x >> 5;  // 0..7
    float s = 0.0f;
    for (int node = blockIdx.x * 8 + rl; node < n; node += gridDim.x * 8)
        s += agg2[(size_t)node * 32 + c];
    __shared__ float red[256];
    red[threadIdx.x] = s;
    __syncthreads();
    if (rl == 0) {
        float t = 0.0f;
#pragma unroll
        for (int r = 0; r < 8; ++r) t += red[r * 32 + c];
        atomicAdd(&csum[c], t);
    }
}

__global__ void finalize32(const float* __restrict__ csum,
                           const float* __restrict__ b2,
                           float* __restrict__ out, float invN) {
    int c = threadIdx.x;
    if (c < 32) out[c] = csum[c] * invN + b2[c];
}

// ------------------------------- launcher ----------------------------------

extern "C" void kernel_launch(void* const* d_in, const int* in_sizes, int n_in,
                              void* d_out, int out_size, void* d_ws,
                              size_t ws_size, hipStream_t stream) {
    const float*     x  = (const float*)d_in[0];
    const long long* ei = (const long long*)d_in[1];  // int64 [2, E]
    const float*     W1 = (const float*)d_in[2];
    const float*     b1 = (const float*)d_in[3];
    const float*     W2 = (const float*)d_in[4];
    const float*     b2 = (const float*)d_in[5];
    float*           out = (float*)d_out;

    const int N = in_sizes[0] / 128;  // 50000
    const int E = in_sizes[1] / 2;    // 800000
    const long long* srcIdx = ei;
    const long long* dstIdx = ei + E;

    // Workspace carve-out (256B aligned slices).
    char* ws = (char*)d_ws;
    size_t off = 0;
    auto carve = [&](size_t bytes) {
        size_t o = off;
        off = (off + bytes + 255) & ~(size_t)255;
        return o;
    };
    float* deg  = (float*)(ws + carve((size_t)N * 4));        // -> dinv in place
    float* h1   = (float*)(ws + carve((size_t)N * 64 * 4));
    float* agg1 = (float*)(ws + carve((size_t)N * 64 * 4));
    float* h2   = (float*)(ws + carve((size_t)N * 32 * 4));
    float* agg2 = (float*)(ws + carve((size_t)N * 32 * 4));
    float* csum = (float*)(ws + carve(128));
    float* z = h1;  // h1 dead after aggregation1; reuse for relu output

    const int TB = 256;
    const long long n64 = (long long)N * 64;
    const long long n32 = (long long)N * 32;

    // --- per-call (re)initialization: graph-replay safe ---
    set_f32<<<(N + TB - 1) / TB, TB, 0, stream>>>(deg, 1.0f, N);  // self-loop
    zero_f32<<<(int)((n64 + TB - 1) / TB), TB, 0, stream>>>(agg1, n64);
    zero_f32<<<(int)((n32 + TB - 1) / TB), TB, 0, stream>>>(agg2, n32);
    zero_f32<<<1, 32, 0, stream>>>(csum, 32);

    // --- degrees & symmetric normalization ---
    count_deg<<<(E + TB - 1) / TB, TB, 0, stream>>>(dstIdx, deg, E);
    deg_to_dinv<<<(N + TB - 1) / TB, TB, 0, stream>>>(deg, N);  // deg == dinv now

    // --- layer 1 ---
    gcn_gemm_wmma<128, 64><<<(N + 127) / 128, 256, 0, stream>>>(x, W1, h1, N);
    gcn_scatter_edges<64>
        <<<(int)(((long long)E * 16 + TB - 1) / TB), TB, 0, stream>>>(
            srcIdx, dstIdx, deg, h1, agg1, E);
    gcn_scatter_self<64>
        <<<(int)(((long long)N * 16 + TB - 1) / TB), TB, 0, stream>>>(
            deg, h1, agg1, N);
    bias_relu64<<<(int)((n64 + TB - 1) / TB), TB, 0, stream>>>(agg1, b1, z, n64);

    // --- layer 2 ---
    gcn_gemm_wmma<64, 32><<<(N + 127) / 128, 256, 0, stream>>>(z, W2, h2, N);
    gcn_scatter_edges<32>
        <<<(int)(((long long)E * 8 + TB - 1) / TB), TB, 0, stream>>>(
            srcIdx, dstIdx, deg, h2, agg2, E);
    gcn_scatter_self<32>
        <<<(int)(((long long)N * 8 + TB - 1) / TB), TB, 0, stream>>>(
            deg, h2, agg2, N);

    // --- global mean pool + bias ---
    colsum32<<<256, 256, 0, stream>>>(agg2, csum, N);
    finalize32<<<1, 32, 0, stream>>>(csum, b2, out, 1.0f / (float)N);
}